// CRFLoss_69200513073848
// MI455X (gfx1250) — compile-verified
//
#include <hip/hip_runtime.h>

typedef float v2f __attribute__((ext_vector_type(2)));
typedef float v8f __attribute__((ext_vector_type(8)));

#define NEGF (-1e30f)

namespace {
constexpr int Csym = 128;   // L+3 symbol classes
constexpr int Bb   = 64;
constexpr int Tt   = 4096;

constexpr int TLEN = 16;                        // timesteps per WMMA tile
constexpr int TPW  = 2;                         // tiles per wave
constexpr int GROUP_T = TPW * TLEN;             // 32 timesteps per wave
constexpr int GROUPS_PER_B = Tt / GROUP_T;      // 128
constexpr int NWAVES = Bb * GROUPS_PER_B;       // 8192
constexpr int BLOCK  = 256;                     // 8 wave32 per block
constexpr int WAVES_PER_BLOCK = BLOCK / 32;
constexpr int NBLOCKS = NWAVES / WAVES_PER_BLOCK; // 1024
constexpr int CHUNKS = 4;                       // phase-2 chunks per batch
constexpr int CHUNK_G = GROUPS_PER_B / CHUNKS;  // 32
}

__device__ __forceinline__ float lse2(float x, float y) {
    float m = fmaxf(x, y);
    return m + __logf(__expf(x - m) + __expf(y - m));
}

// ---------------------------------------------------------------------------
// Kernel 1 (one 256-thread block): per-state log_softmax of arc scores ->
// prob-domain weight tables + pre-swizzled WMMA B-fragment table.
//   wtab[0:128)   = w0[c], wtab[128:256) = w1[c]
//   Btab[(j*32+lane)*2 .. +1] = B-fragment float2 for K-step j, lane
//   scal[0] = s0[126] ('I-' 0->1)  scal[1] = s1[0] ('I-' 1->1)  scal[2] = s0[127]
// ---------------------------------------------------------------------------
__global__ void __launch_bounds__(256) crf_setup_kernel(
        const float* __restrict__ den,
        float* __restrict__ wtab,
        float* __restrict__ Btab,
        float* __restrict__ scal) {
    __shared__ float red[256];
    __shared__ float sw0[128], sw1[128];
    const int tid = threadIdx.x;

    // ---- log-normalizer, state 0 (128 arcs) ----
    red[tid] = (tid < 128) ? den[tid] : NEGF;
    __syncthreads();
    for (int off = 128; off >= 1; off >>= 1) {
        if (tid < off) red[tid] = fmaxf(red[tid], red[tid + off]);
        __syncthreads();
    }
    const float mx0 = red[0];
    __syncthreads();
    red[tid] = (tid < 128) ? expf(den[tid] - mx0) : 0.f;
    __syncthreads();
    for (int off = 128; off >= 1; off >>= 1) {
        if (tid < off) red[tid] += red[tid + off];
        __syncthreads();
    }
    const float lz0 = mx0 + logf(red[0]);
    __syncthreads();

    // ---- log-normalizer, state 1 (126 arcs) ----
    red[tid] = (tid < 126) ? den[128 + tid] : NEGF;
    __syncthreads();
    for (int off = 128; off >= 1; off >>= 1) {
        if (tid < off) red[tid] = fmaxf(red[tid], red[tid + off]);
        __syncthreads();
    }
    const float mx1 = red[0];
    __syncthreads();
    red[tid] = (tid < 126) ? expf(den[128 + tid] - mx1) : 0.f;
    __syncthreads();
    for (int off = 128; off >= 1; off >>= 1) {
        if (tid < off) red[tid] += red[tid + off];
        __syncthreads();
    }
    const float lz1 = mx1 + logf(red[0]);
    __syncthreads();

    // ---- weight tables in prob domain, scattered onto symbol ids ----
    if (tid < 128) {
        const int c = tid;
        float w0v = 0.f, w1v = 0.f;
        if (c == 1) w0v = expf(den[0] - lz0);               // 'O' arc (label 1)
        else if (c >= 3) w0v = expf(den[c - 2] - lz0);      // labels 3..127, 0->0
        if (c >= 3) w1v = expf(den[128 + (c - 2)] - lz1);   // labels 3..127, 1->0
        sw0[c] = w0v; sw1[c] = w1v;
        wtab[c] = w0v; wtab[128 + c] = w1v;
    }
    __syncthreads();

    // ---- pre-swizzled B fragments: lane n=lane&15 carries col n (w0/w1/zero)
    for (int e = tid; e < 32 * 32; e += 256) {
        const int j = e >> 5, lane = e & 31;
        const int n = lane & 15;
        const int kk = (lane < 16) ? 0 : 2;
        const int k = 4 * j + kk;
        float b0 = 0.f, b1 = 0.f;
        if (n == 0)      { b0 = sw0[k]; b1 = sw0[k + 1]; }
        else if (n == 1) { b0 = sw1[k]; b1 = sw1[k + 1]; }
        Btab[2 * e]     = b0;
        Btab[2 * e + 1] = b1;
    }
    if (tid == 0) {
        scal[0] = den[126] - lz0;   // M01 const
        scal[1] = den[128] - lz1;   // M11 const
        scal[2] = den[127] - lz0;   // final score
    }
}

// ---------------------------------------------------------------------------
// Kernel 2: one wave per 32 consecutive timesteps of one batch row.
//   Dual-accumulator WMMA f32 16x16x4 chain (K=128) computes the two
//   prob-domain dots for 16 timesteps; lane 0 folds the per-t 2x2
//   log-semiring matrices in time order into a per-wave partial product.
// ---------------------------------------------------------------------------
__global__ void __launch_bounds__(BLOCK) crf_phase1_kernel(
        const float* __restrict__ lp,
        const int*   __restrict__ lens,
        const int*   __restrict__ labels,
        const float* __restrict__ Btab,
        const float* __restrict__ scal,
        float*       __restrict__ Mtab,
        float*       __restrict__ numpart)
{
    __shared__ float lds[WAVES_PER_BLOCK][TLEN][4];  // [t][ M00, M10, M01, M11 ]
    const int wib  = threadIdx.x >> 5;
    const int lane = threadIdx.x & 31;
    const int wave = blockIdx.x * WAVES_PER_BLOCK + wib;
    const int b = wave / GROUPS_PER_B;
    const int g = wave % GROUPS_PER_B;
    const int len = lens[b];

    const int  m  = lane & 15;
    const bool hi = lane >= 16;
    const int  kk = hi ? 2 : 0;
    const float a10c = scal[0];
    const float a11c = scal[1];
    const v2f* __restrict__ Bf = ((const v2f*)Btab) + lane;   // Bf[j*32]

    float p00 = 0.f, p01 = NEGF, p10 = NEGF, p11 = 0.f; // running 2x2 product
    float numacc = 0.f;

    for (int ti = 0; ti < TPW; ++ti) {
        const int t0 = g * GROUP_T + ti * TLEN;
        // A-frag row for this lane: timestep t0+m (lanes 16..31 mirror rows 0..15)
        const float* rowp = lp + ((size_t)b * Tt + (size_t)(t0 + m)) * Csym;

        v8f acc0 = {0.f, 0.f, 0.f, 0.f, 0.f, 0.f, 0.f, 0.f};
        v8f acc1 = {0.f, 0.f, 0.f, 0.f, 0.f, 0.f, 0.f, 0.f};
        #pragma unroll 4
        for (int j = 0; j < 32; j += 2) {
            const int k0 = 4 * j;
            v2f a0;
            a0.x = __expf(rowp[k0 + kk]);
            a0.y = __expf(rowp[k0 + kk + 1]);
            const v2f b0 = Bf[j * 32];
            acc0 = __builtin_amdgcn_wmma_f32_16x16x4_f32(
                       false, a0, false, b0, (short)0, acc0, false, false);
            v2f a1;
            a1.x = __expf(rowp[k0 + 4 + kk]);
            a1.y = __expf(rowp[k0 + 4 + kk + 1]);
            const v2f b1 = Bf[(j + 1) * 32];
            acc1 = __builtin_amdgcn_wmma_f32_16x16x4_f32(
                       false, a1, false, b1, (short)0, acc1, false, false);
        }
        const v8f acc = acc0 + acc1;

        // D layout: VGPR r, lanes 0-15 -> row r, lanes 16-31 -> row r+8; col = lane&15
        if (m < 2) {
            #pragma unroll
            for (int r = 0; r < 8; ++r) {
                const int tt = r + (hi ? 8 : 0);
                lds[wib][tt][m] = __logf(fmaxf(acc[r], 1e-37f)); // col0->M00, col1->M10
            }
        }
        if (!hi) {
            const float lp2 = rowp[2];              // row t0+lane, symbol 'I-'
            lds[wib][lane][2] = a10c + lp2;         // M01
            lds[wib][lane][3] = a11c + lp2;         // M11
            const int t = t0 + lane;                // numerator gather
            if (t < len) {
                const int lb = labels[(size_t)b * Tt + t];
                numacc += lp[((size_t)b * Tt + t) * Csym + lb];
            }
        }
        __syncthreads();
        if (lane == 0) {
            for (int i = 0; i < TLEN; ++i) {
                const int t = t0 + i;
                if (t < len) {                      // t >= len: identity, skip
                    const float m00 = lds[wib][i][0];
                    const float m10 = lds[wib][i][1];
                    const float m01 = lds[wib][i][2];
                    const float m11 = lds[wib][i][3];
                    const float n00 = lse2(p00 + m00, p01 + m10);
                    const float n01 = lse2(p00 + m01, p01 + m11);
                    const float n10 = lse2(p10 + m00, p11 + m10);
                    const float n11 = lse2(p10 + m01, p11 + m11);
                    p00 = n00; p01 = n01; p10 = n10; p11 = n11;
                }
            }
        }
        __syncthreads();
    }

    // numerator: reduce lanes 0..15 (hi lanes hold 0)
    numacc += __shfl_down(numacc, 8, 32);
    numacc += __shfl_down(numacc, 4, 32);
    numacc += __shfl_down(numacc, 2, 32);
    numacc += __shfl_down(numacc, 1, 32);
    if (lane == 0) {
        float4 o; o.x = p00; o.y = p01; o.z = p10; o.w = p11;
        ((float4*)Mtab)[wave] = o;
        numpart[wave] = numacc;
    }
}

// ---------------------------------------------------------------------------
// Kernel 3 (one 256-thread block): two-stage ordered fold of per-wave 2x2
// partials (4 chunks x 32 groups per batch), double-precision totals.
// ---------------------------------------------------------------------------
__global__ void __launch_bounds__(256) crf_phase2_kernel(
        const float*  __restrict__ Mtab,
        const float*  __restrict__ numpart,
        const float*  __restrict__ scal,
        float*        __restrict__ out)
{
    __shared__ float  chunk[Bb * CHUNKS][4];
    __shared__ double csum[Bb * CHUNKS];
    __shared__ double dden[Bb];
    __shared__ double dnum[Bb];
    const int tid = threadIdx.x;        // 256 = 64 batches x 4 chunks
    const int b = tid >> 2;
    const int q = tid & 3;

    float p00 = 0.f, p01 = NEGF, p10 = NEGF, p11 = 0.f;
    double ns = 0.0;
    const int base = b * GROUPS_PER_B + q * CHUNK_G;
    for (int i = 0; i < CHUNK_G; ++i) {
        const float4 qm = ((const float4*)Mtab)[base + i];
        const float n00 = lse2(p00 + qm.x, p01 + qm.z);
        const float n01 = lse2(p00 + qm.y, p01 + qm.w);
        const float n10 = lse2(p10 + qm.x, p11 + qm.z);
        const float n11 = lse2(p10 + qm.y, p11 + qm.w);
        p00 = n00; p01 = n01; p10 = n10; p11 = n11;
        ns += (double)numpart[base + i];
    }
    chunk[tid][0] = p00; chunk[tid][1] = p01;
    chunk[tid][2] = p10; chunk[tid][3] = p11;
    csum[tid] = ns;
    __syncthreads();

    if (tid < Bb) {
        float r00 = 0.f, r01 = NEGF, r10 = NEGF, r11 = 0.f;
        double nb = 0.0;
        for (int k = 0; k < CHUNKS; ++k) {
            const int e = tid * CHUNKS + k;
            const float q00 = chunk[e][0], q01 = chunk[e][1];
            const float q10 = chunk[e][2], q11 = chunk[e][3];
            const float n00 = lse2(r00 + q00, r01 + q10);
            const float n01 = lse2(r00 + q01, r01 + q11);
            const float n10 = lse2(r10 + q00, r11 + q10);
            const float n11 = lse2(r10 + q01, r11 + q11);
            r00 = n00; r01 = n01; r10 = n10; r11 = n11;
            nb += csum[e];
        }
        dden[tid] = (double)(r00 + scal[2]);   // alpha_T[0] + final (alpha_0=[0,-inf])
        dnum[tid] = nb;
    }
    __syncthreads();
    if (tid == 0) {
        double den = 0.0, num = 0.0;
        for (int i = 0; i < Bb; ++i) { den += dden[i]; num += dnum[i]; }
        out[0] = (float)(num - den);
    }
}

extern "C" void kernel_launch(void* const* d_in, const int* in_sizes, int n_in,
                              void* d_out, int out_size, void* d_ws, size_t ws_size,
                              hipStream_t stream)
{
    (void)in_sizes; (void)n_in; (void)out_size; (void)ws_size;
    const float* log_probs  = (const float*)d_in[0];
    const float* den_scores = (const float*)d_in[1];
    const int*   input_lens = (const int*)d_in[2];
    const int*   labels     = (const int*)d_in[3];
    float* out = (float*)d_out;

    char* ws = (char*)d_ws;
    float* scal = (float*)(ws);                         //   16 B (3 floats used)
    float* wtab = (float*)(ws + 16);                    // 1024 B (w0[128], w1[128])
    float* Btab = (float*)(ws + 16 + 1024);             // 8192 B (32 j x 32 lanes x v2f)
    float* Mtab = (float*)(ws + 16 + 1024 + 8192);      // 8192 * 16 B = 128 KB
    float* numpart = (float*)(ws + 16 + 1024 + 8192 + (size_t)NWAVES * 16); // 32 KB

    crf_setup_kernel <<<1, 256, 0, stream>>>(den_scores, wtab, Btab, scal);
    crf_phase1_kernel<<<NBLOCKS, BLOCK, 0, stream>>>(log_probs, input_lens, labels,
                                                     Btab, scal, Mtab, numpart);
    crf_phase2_kernel<<<1, 256, 0, stream>>>(Mtab, numpart, scal, out);
}